// MetroGNN_43731357008588
// MI455X (gfx1250) — compile-verified
//
#include <hip/hip_runtime.h>

typedef __attribute__((ext_vector_type(2))) float v2f;
typedef __attribute__((ext_vector_type(8))) float v8f;

#define NN 200000
#define NE 6400000
#define NTILES (NN / 16)   // 12500, exact

// ---- init: deg=1 (self-loop weight), zero aggx[3N], zero out[4N] -----------
__global__ void k_init(float* __restrict__ deg, float* __restrict__ aggx,
                       float* __restrict__ out2) {
    int i = blockIdx.x * blockDim.x + threadIdx.x;   // covers 4*NN
    if (i < NN) deg[i] = 1.0f;
    if (i < 3 * NN) aggx[i] = 0.0f;
    if (i < 4 * NN) out2[i] = 0.0f;
}

// ---- weighted in-degree at targets -----------------------------------------
__global__ void k_deg(const int* __restrict__ col, const float* __restrict__ w,
                      float* __restrict__ deg) {
    int e = blockIdx.x * blockDim.x + threadIdx.x;
    if (e < NE) atomicAdd(&deg[col[e]], w[e]);
}

// ---- dis = deg>0 ? rsqrt(deg) : 0 (in place) -------------------------------
__global__ void k_dis(float* __restrict__ deg) {
    int i = blockIdx.x * blockDim.x + threadIdx.x;
    if (i < NN) {
        float d = deg[i];
        deg[i] = (d > 0.0f) ? rsqrtf(d) : 0.0f;
    }
}

// ---- layer-1 raw-feature scatter: aggx[c] += x[r]*norm (3 atomics/edge) ----
// Valid because segsum((x[row]@W1)*norm) == (segsum(x[row]*norm))@W1.
__global__ void k_scatter1(const int* __restrict__ row, const int* __restrict__ col,
                           const float* __restrict__ w, const float* __restrict__ dis,
                           const float* __restrict__ x, float* __restrict__ aggx) {
    int e = blockIdx.x * blockDim.x + threadIdx.x;
    if (e >= NE) return;
    int r = row[e], c = col[e];
    float nrm = dis[r] * w[e] * dis[c];
    const float* xp = x + (size_t)r * 3;
    float* op = aggx + (size_t)c * 3;
    atomicAdd(op + 0, xp[0] * nrm);
    atomicAdd(op + 1, xp[1] * nrm);
    atomicAdd(op + 2, xp[2] * nrm);
}

// ---- fused: z=(aggx+x*dis^2); h1=relu(z@W1+b1) [WMMA]; LDS transpose;
//      h2 = h1@W2 [4x chained WMMA] --------------------------------------
__global__ void k_l1l2(const float* __restrict__ x, const float* __restrict__ aggx,
                       const float* __restrict__ dis, const float* __restrict__ W1,
                       const float* __restrict__ b1, const float* __restrict__ W2,
                       float* __restrict__ h2) {
    __shared__ float lds[8 * 256];                       // 1KB tile per wave
    int tile = (blockIdx.x * blockDim.x + threadIdx.x) >> 5;
    int lane = threadIdx.x & 31;
    float* T = &lds[(threadIdx.x >> 5) * 256];           // wave-private [k][m]
    if (tile >= NTILES) return;                          // wave-uniform exit

    int m = lane & 15;
    bool hi = lane >= 16;                                // lanes 16-31: K={2,3}
    int node = tile * 16 + m;
    int n = m;

    // ---- layer 1: A = z[node][0..3], B = W1[4(pad) x 16] ----
    float ds0 = dis[node];
    float s = ds0 * ds0;                                 // self-loop norm
    const float* xp = x + (size_t)node * 3;
    const float* ap = aggx + (size_t)node * 3;
    v2f a, b;
    if (!hi) { a.x = ap[0] + xp[0] * s; a.y = ap[1] + xp[1] * s; }  // K=0,1
    else     { a.x = ap[2] + xp[2] * s; a.y = 0.0f; }               // K=2, pad
    if (!hi) { b.x = W1[0 * 16 + n]; b.y = W1[1 * 16 + n]; }
    else     { b.x = W1[2 * 16 + n]; b.y = 0.0f; }

    v8f c1 = {};
    c1 = __builtin_amdgcn_wmma_f32_16x16x4_f32(false, a, false, b, (short)0, c1,
                                               false, false);

    // bias + relu, store transposed: T[feature][node] so layer-2 A reads rows
    int mBase = hi ? 8 : 0;
    float bias = b1[n];
#pragma unroll
    for (int r = 0; r < 8; ++r)
        T[n * 16 + mBase + r] = fmaxf(c1[r] + bias, 0.0f);

    // ---- layer 2: K=16 in 4 chunks of 4, A from LDS, pad output cols ----
    v8f c2 = {};
#pragma unroll
    for (int kc = 0; kc < 4; ++kc) {
        int k0 = 4 * kc + (hi ? 2 : 0);
        v2f a2, b2v;
        a2.x = T[k0 * 16 + m];
        a2.y = T[(k0 + 1) * 16 + m];
        b2v.x = (n < 4) ? W2[k0 * 4 + n] : 0.0f;
        b2v.y = (n < 4) ? W2[(k0 + 1) * 4 + n] : 0.0f;
        c2 = __builtin_amdgcn_wmma_f32_16x16x4_f32(false, a2, false, b2v, (short)0,
                                                   c2, false, false);
    }
    if (n < 4) {
#pragma unroll
        for (int r = 0; r < 8; ++r)
            h2[(size_t)(tile * 16 + mBase + r) * 4 + n] = c2[r];
    }
}

// ---- layer-2 edge scatter into d_out (4 atomics/edge) ----------------------
__global__ void k_scatter2(const int* __restrict__ row, const int* __restrict__ col,
                           const float* __restrict__ w, const float* __restrict__ dis,
                           const float* __restrict__ h2, float* __restrict__ out) {
    int e = blockIdx.x * blockDim.x + threadIdx.x;
    if (e >= NE) return;
    int r = row[e], c = col[e];
    float nrm = dis[r] * w[e] * dis[c];
    const float4 hv = *(const float4*)(h2 + (size_t)r * 4);
    float* op = out + (size_t)c * 4;
    atomicAdd(op + 0, hv.x * nrm);
    atomicAdd(op + 1, hv.y * nrm);
    atomicAdd(op + 2, hv.z * nrm);
    atomicAdd(op + 3, hv.w * nrm);
}

// ---- finalize layer 2: out += h2*dis^2 + b2 --------------------------------
__global__ void k_fin2(const float* __restrict__ h2, const float* __restrict__ dis,
                       const float* __restrict__ b2, float* __restrict__ out) {
    int idx = blockIdx.x * blockDim.x + threadIdx.x;
    if (idx >= 4 * NN) return;
    int i = idx >> 2, k = idx & 3;
    float d = dis[i];
    out[idx] += h2[idx] * d * d + b2[k];
}

extern "C" void kernel_launch(void* const* d_in, const int* in_sizes, int n_in,
                              void* d_out, int out_size, void* d_ws, size_t ws_size,
                              hipStream_t stream) {
    const float* x   = (const float*)d_in[0];
    const int*   ei  = (const int*)d_in[1];     // [2, NE] int32
    const float* ew  = (const float*)d_in[2];
    const float* W1  = (const float*)d_in[3];
    const float* b1  = (const float*)d_in[4];
    const float* W2  = (const float*)d_in[5];
    const float* b2  = (const float*)d_in[6];
    float* out = (float*)d_out;                 // [NN, 4]

    const int* row = ei;
    const int* col = ei + NE;

    // workspace: dis[N] | aggx[3N] | h2[4N]  => 8N floats (~6.4 MB)
    float* dis  = (float*)d_ws;
    float* aggx = dis + NN;
    float* h2   = aggx + (size_t)3 * NN;

    const int B = 256;
    const int gInit = (4 * NN + B - 1) / B;
    const int gEdge = (NE + B - 1) / B;
    const int gNode = (NN + B - 1) / B;
    const int gTile = (NTILES + (B / 32) - 1) / (B / 32);  // 8 waves/block
    const int gF2   = (4 * NN + B - 1) / B;

    k_init    <<<gInit, B, 0, stream>>>(dis, aggx, out);
    k_deg     <<<gEdge, B, 0, stream>>>(col, ew, dis);
    k_dis     <<<gNode, B, 0, stream>>>(dis);
    k_scatter1<<<gEdge, B, 0, stream>>>(row, col, ew, dis, x, aggx);
    k_l1l2    <<<gTile, B, 0, stream>>>(x, aggx, dis, W1, b1, W2, h2);
    k_scatter2<<<gEdge, B, 0, stream>>>(row, col, ew, dis, h2, out);
    k_fin2    <<<gF2,   B, 0, stream>>>(h2, dis, b2, out);
}